// PerSquareInputEncoder_35682588295306
// MI455X (gfx1250) — compile-verified
//
#include <hip/hip_runtime.h>

typedef float v2f __attribute__((ext_vector_type(2)));
typedef float v4f __attribute__((ext_vector_type(4)));
typedef float v8f __attribute__((ext_vector_type(8)));

#define B_SZ 4096
#define D_SZ 256

// ---------------------------------------------------------------------------
// Kernel 1: meta_vec(B,256) = meta(B,28) @ W_meta(28,256) via f32 WMMA.
// meta = concat(time_history[8], rep_flags[8], castling[4], scalars[8]).
// One wave (32 lanes) per 16x16 output tile; K=28 -> 7 steps of K=4.
// Exact f32 math: V_WMMA_F32_16X16X4_F32 accumulates in f32 from f32 inputs.
// The 16x28 meta tile is staged in LDS with uniform (divergence-free) copy
// loops so the WMMA inner loop is just ds_load + global_load + v_wmma.
// ---------------------------------------------------------------------------
__global__ __launch_bounds__(32) void meta_wmma_kernel(
    const float* __restrict__ time_history,  // (B,8)
    const float* __restrict__ rep_flags,     // (B,8)
    const float* __restrict__ castling,      // (B,4)
    const float* __restrict__ scalars,       // (B,8)
    const float* __restrict__ W_meta,        // (28,256)
    float* __restrict__ meta_vec)            // (B,256)
{
    __shared__ float sMeta[16][28];          // one 16-row x 28-feature tile

    const int lane    = threadIdx.x;         // 0..31
    const int m16     = lane & 15;
    const int hi      = lane >> 4;           // 0: K 0..1, 1: K 2..3 (f32 A layout)
    const int rowBase = blockIdx.x * 16;     // tile over batch dimension
    const int colBase = blockIdx.y * 16;     // tile over D dimension

    // ---- Stage meta tile into LDS: four uniform loops, no lane divergence.
#pragma unroll
    for (int i = lane; i < 16 * 8; i += 32)  // time_history -> cols 0..7
        sMeta[i >> 3][i & 7] = time_history[(rowBase + (i >> 3)) * 8 + (i & 7)];
#pragma unroll
    for (int i = lane; i < 16 * 8; i += 32)  // rep_flags -> cols 8..15
        sMeta[i >> 3][8 + (i & 7)] = rep_flags[(rowBase + (i >> 3)) * 8 + (i & 7)];
#pragma unroll
    for (int i = lane; i < 16 * 4; i += 32)  // castling -> cols 16..19
        sMeta[i >> 2][16 + (i & 3)] = castling[(rowBase + (i >> 2)) * 4 + (i & 3)];
#pragma unroll
    for (int i = lane; i < 16 * 8; i += 32)  // scalars -> cols 20..27
        sMeta[i >> 3][20 + (i & 7)] = scalars[(rowBase + (i >> 3)) * 8 + (i & 7)];
    __syncthreads();

    v8f c = {};
#pragma unroll
    for (int kb = 0; kb < 28; kb += 4) {
        v2f a, w;
#pragma unroll
        for (int r = 0; r < 2; ++r) {
            const int f = kb + r + (hi << 1);        // this lane's K index
            a[r] = sMeta[m16][f];                    // A 16x4: lane m16 holds row m16
            w[r] = W_meta[f * D_SZ + colBase + m16]; // B 4x16: row K striped across lanes
        }
        c = __builtin_amdgcn_wmma_f32_16x16x4_f32(
            /*neg_a=*/false, a, /*neg_b=*/false, w,
            /*c_mod=*/(short)0, c, /*reuse_a=*/false, /*reuse_b=*/false);
    }

    // C/D 16x16 f32 layout: VGPR r -> M = r (lanes 0-15) or r+8 (lanes 16-31)
#pragma unroll
    for (int r = 0; r < 8; ++r) {
        const int row = rowBase + r + (hi ? 8 : 0);
        meta_vec[row * D_SZ + colBase + m16] = c[r];
    }
}

// ---------------------------------------------------------------------------
// Kernel 2: fused one-hot gather + affine + embeddings.
// Block = 256 threads = 4 groups of 64; each group produces one (b,s) row of
// 256 floats via float4 (b128) accesses. Piece codes staged via LDS so each
// block does the 8 code loads once per row. Output stored non-temporally
// (256 MB stream; keep L2 for the hot W_board / per-square tables).
// ---------------------------------------------------------------------------
__global__ __launch_bounds__(256) void encode_kernel(
    const int*   __restrict__ board_history, // (B,8,64)
    const float* __restrict__ ep_mask,       // (B,64)
    const int*   __restrict__ tc_cat,        // (B)
    const float* __restrict__ W_board,       // (105,256)
    const float* __restrict__ square_emb,    // (64,256)
    const float* __restrict__ square_scale,  // (64,256)
    const float* __restrict__ square_bias,   // (64,256)
    const float* __restrict__ tc_table,      // (8,256)
    const float* __restrict__ meta_vec,      // (B,256)
    float*       __restrict__ out)           // (B,64,256)
{
    __shared__ int   sCodes[4][8];
    __shared__ float sEp[4];

    const int t    = threadIdx.x;
    const int g    = t >> 6;                 // group 0..3
    const int l    = t & 63;                 // lane within group
    const int pair = blockIdx.x * 4 + g;     // flat (b,s) index
    const int b    = pair >> 6;
    const int s    = pair & 63;

    // Stage the 8 history codes + ep flag for each of the 4 rows.
    if (t < 32) {
        const int gg = t >> 3, h = t & 7;
        const int pb = blockIdx.x * 4 + gg;
        const int bb = pb >> 6, ss = pb & 63;
        sCodes[gg][h] = board_history[(bb * 8 + h) * 64 + ss];
    } else if (t < 36) {
        const int gg = t - 32;
        sEp[gg] = ep_mask[blockIdx.x * 4 + gg];
    }
    __syncthreads();

    const int d0 = l * 4;                    // this thread's 4 channels

    v4f acc = *(const v4f*)(meta_vec + b * D_SZ + d0);
    acc += *(const v4f*)(square_emb + s * D_SZ + d0);

#pragma unroll
    for (int h = 0; h < 8; ++h) {
        const int code = sCodes[g][h];
        acc += *(const v4f*)(W_board + (h * 13 + code) * D_SZ + d0);
    }

    const float ep = sEp[g];
    const v4f wep = *(const v4f*)(W_board + 104 * D_SZ + d0);
    acc += ep * wep;

    const v4f sc = *(const v4f*)(square_scale + s * D_SZ + d0);
    const v4f bi = *(const v4f*)(square_bias + s * D_SZ + d0);
    acc = acc * sc + bi;

    const int tc = tc_cat[b];
    acc += *(const v4f*)(tc_table + tc * D_SZ + d0);

    __builtin_nontemporal_store(acc, (v4f*)(out + (size_t)pair * D_SZ + d0));
}

// ---------------------------------------------------------------------------
// Launch
// ---------------------------------------------------------------------------
extern "C" void kernel_launch(void* const* d_in, const int* in_sizes, int n_in,
                              void* d_out, int out_size, void* d_ws, size_t ws_size,
                              hipStream_t stream) {
    const int*   board_history = (const int*)  d_in[0];
    const float* time_history  = (const float*)d_in[1];
    const float* rep_flags     = (const float*)d_in[2];
    const float* castling      = (const float*)d_in[3];
    const float* ep_mask       = (const float*)d_in[4];
    const float* scalars       = (const float*)d_in[5];
    const int*   tc_cat        = (const int*)  d_in[6];
    const float* W_board       = (const float*)d_in[7];
    const float* W_meta        = (const float*)d_in[8];
    const float* square_emb    = (const float*)d_in[9];
    const float* square_scale  = (const float*)d_in[10];
    const float* square_bias   = (const float*)d_in[11];
    const float* tc_table      = (const float*)d_in[12];

    float* meta_vec = (float*)d_ws;          // B * D floats = 4 MB
    float* out      = (float*)d_out;         // (B, 64, D)

    // Kernel 1: 256 M-tiles (B/16) x 16 N-tiles (D/16), one wave each.
    dim3 grid1(B_SZ / 16, D_SZ / 16);
    meta_wmma_kernel<<<grid1, 32, 0, stream>>>(
        time_history, rep_flags, castling, scalars, W_meta, meta_vec);

    // Kernel 2: 4 (b,s) rows per 256-thread block.
    dim3 grid2((B_SZ * 64) / 4);
    encode_kernel<<<grid2, 256, 0, stream>>>(
        board_history, ep_mask, tc_cat, W_board, square_emb,
        square_scale, square_bias, tc_table, meta_vec, out);
}